// Decoder_55164559950037
// MI455X (gfx1250) — compile-verified
//
#include <hip/hip_runtime.h>
#include <hip/hip_bf16.h>
#include <math.h>

typedef __attribute__((ext_vector_type(2))) float v2f;
typedef __attribute__((ext_vector_type(8))) float v8f;

#define B_SZ 512
#define E_SZ 512
#define W_SZ 64
#define V_SZ 32000
#define H_SZ (10 * E_SZ)   // 5120

// ---------------------------------------------------------------------------
// fp32 WMMA GEMM:  C[M,N] = act(A[M,K] @ Bm[K,N] + bias[N])
// One wave computes a 32x64 strip: 2 M-tiles x 4 N-tiles of 16x16, K stepped
// by 4 with V_WMMA_F32_16X16X4_F32. B fragments are shared by both M-tiles;
// the K-loop is software-pipelined (next step's loads issued before this
// step's 8 WMMAs) and uses running pointers (no per-iter address mul).
// Wave->tile map puts all M-tiles of one N-strip on consecutive waves so
// co-scheduled waves stream each B column from HBM once (L2 reuse).
// Requires: M % 32 == 0, N % 64 == 0, K % 4 == 0.
// ---------------------------------------------------------------------------
template <int ACT>  // 0 = none, 1 = tanh
__global__ __launch_bounds__(256) void gemm_f32_wmma(
    const float* __restrict__ A, const float* __restrict__ Bm,
    const float* __restrict__ bias, float* __restrict__ C,
    int M, int N, int K) {
  const int wave = threadIdx.x >> 5;   // 8 waves / block
  const int lane = threadIdx.x & 31;
  const int mtiles = M >> 5;           // 32-row strips
  const int ntiles = N >> 6;           // 64-col strips
  const int wid = blockIdx.x * 8 + wave;
  if (wid >= mtiles * ntiles) return;  // wave-uniform: EXEC stays all-ones
  const int mt = wid % mtiles;         // M varies fastest -> B reuse in L2
  const int nt = wid / mtiles;
  const int m0 = mt << 5;
  const int n0 = nt << 6;

  // ISA fragment mapping (32-bit A 16x4 / B 4x16), lane L:
  //   A: row = base + (L&15); v.x = K+klo, v.y = K+klo+1, klo = (L>>4)*2
  //   B: col = n0 + (L&15) + 16j; v.x = row K+klo, v.y = row K+klo+1
  const int l15 = lane & 15;
  const int klo = (lane >> 4) << 1;    // 0 or 2

  const float* ap0 = A + (size_t)(m0 + l15) * K + klo;        // M-tile 0
  const float* ap1 = ap0 + (size_t)16 * K;                    // M-tile 1
  const float* bp  = Bm + (size_t)klo * N + n0 + l15;
  const size_t bstep = (size_t)4 * N;

  v8f acc0[4] = {};
  v8f acc1[4] = {};

  v2f a0c, a1c, bc[4];
  // prologue load (k = 0)
  a0c.x = ap0[0]; a0c.y = ap0[1];
  a1c.x = ap1[0]; a1c.y = ap1[1];
#pragma unroll
  for (int j = 0; j < 4; ++j) {
    bc[j].x = bp[j * 16];
    bc[j].y = bp[j * 16 + (size_t)N];
  }

  for (int k = 0; k < K - 4; k += 4) {
    ap0 += 4; ap1 += 4; bp += bstep;
    // issue next step's loads first so they overlap the 8 WMMAs below
    v2f a0n, a1n, bn[4];
    a0n.x = ap0[0]; a0n.y = ap0[1];
    a1n.x = ap1[0]; a1n.y = ap1[1];
#pragma unroll
    for (int j = 0; j < 4; ++j) {
      bn[j].x = bp[j * 16];
      bn[j].y = bp[j * 16 + (size_t)N];
    }
#pragma unroll
    for (int j = 0; j < 4; ++j) {
      acc0[j] = __builtin_amdgcn_wmma_f32_16x16x4_f32(
          false, a0c, false, bc[j], (short)0, acc0[j], false, false);
      acc1[j] = __builtin_amdgcn_wmma_f32_16x16x4_f32(
          false, a1c, false, bc[j], (short)0, acc1[j], false, false);
    }
    a0c = a0n; a1c = a1n;
#pragma unroll
    for (int j = 0; j < 4; ++j) bc[j] = bn[j];
  }
  // epilogue
#pragma unroll
  for (int j = 0; j < 4; ++j) {
    acc0[j] = __builtin_amdgcn_wmma_f32_16x16x4_f32(
        false, a0c, false, bc[j], (short)0, acc0[j], false, false);
    acc1[j] = __builtin_amdgcn_wmma_f32_16x16x4_f32(
        false, a1c, false, bc[j], (short)0, acc1[j], false, false);
  }

  // C/D layout: VGPR r -> row base + r + ((lane>>4)<<3), col n0 + 16j + (lane&15)
  const int rhalf = (lane >> 4) << 3;
#pragma unroll
  for (int j = 0; j < 4; ++j) {
    const int c = n0 + j * 16 + l15;
    const float bv = bias ? bias[c] : 0.0f;
#pragma unroll
    for (int r = 0; r < 8; ++r) {
      float v0 = acc0[j][r] + bv;
      float v1 = acc1[j][r] + bv;
      if (ACT == 1) { v0 = tanhf(v0); v1 = tanhf(v1); }
      C[(size_t)(m0 + r + rhalf) * N + c] = v0;
      C[(size_t)(m0 + 16 + r + rhalf) * N + c] = v1;
    }
  }
}

static inline void launch_gemm(const float* A, const float* Bm, const float* bias,
                               float* C, int M, int N, int K, int act,
                               hipStream_t s) {
  const int waves = (M / 32) * (N / 64);
  const int blocks = (waves + 7) / 8;
  if (act)
    gemm_f32_wmma<1><<<blocks, 256, 0, s>>>(A, Bm, bias, C, M, N, K);
  else
    gemm_f32_wmma<0><<<blocks, 256, 0, s>>>(A, Bm, bias, C, M, N, K);
}

// ---------------------------------------------------------------------------
// Attention: one block per batch row b.
// scores[w] = sum_e tanh(state[b,e]*aW1[e] * encode[b,w,e]*aW2[w,e]) * aW3[w,e]
// a = softmax(scores); context[b,e] = sum_w a[w] * encode[b,w,e]
// ---------------------------------------------------------------------------
__global__ __launch_bounds__(256) void attention_kernel(
    const float* __restrict__ state, const float* __restrict__ encode,
    const float* __restrict__ aW1, const float* __restrict__ aW2,
    const float* __restrict__ aW3, float* __restrict__ context) {
  const int b = blockIdx.x;
  const int lane = threadIdx.x & 31;
  const int wave = threadIdx.x >> 5;
  __shared__ float sc[W_SZ];
  __shared__ float atn[W_SZ];

  for (int w = wave; w < W_SZ; w += 8) {
    float s = 0.0f;
    const float* ep = encode + ((size_t)b * W_SZ + w) * E_SZ;
    const float* w2 = aW2 + (size_t)w * E_SZ;
    const float* w3 = aW3 + (size_t)w * E_SZ;
    for (int e = lane; e < E_SZ; e += 32) {
      float rq = state[(size_t)b * E_SZ + e] * aW1[e];
      float rv = ep[e] * w2[e];
      s += tanhf(rq * rv) * w3[e];
    }
#pragma unroll
    for (int o = 16; o; o >>= 1) s += __shfl_xor(s, o, 32);
    if (lane == 0) sc[w] = s;
  }
  __syncthreads();

  if (wave == 0) {  // softmax over 64 scores, 2 per lane
    float v0 = sc[lane], v1 = sc[lane + 32];
    float m = fmaxf(v0, v1);
#pragma unroll
    for (int o = 16; o; o >>= 1) m = fmaxf(m, __shfl_xor(m, o, 32));
    float e0 = __expf(v0 - m), e1 = __expf(v1 - m);
    float s = e0 + e1;
#pragma unroll
    for (int o = 16; o; o >>= 1) s += __shfl_xor(s, o, 32);
    float inv = 1.0f / s;
    atn[lane] = e0 * inv;
    atn[lane + 32] = e1 * inv;
  }
  __syncthreads();

  for (int e = threadIdx.x; e < E_SZ; e += 256) {
    float acc = 0.0f;
    const float* ep = encode + (size_t)b * W_SZ * E_SZ + e;
    for (int w = 0; w < W_SZ; ++w) acc += atn[w] * ep[(size_t)w * E_SZ];
    context[(size_t)b * E_SZ + e] = acc;
  }
}

// seq = stack([state, context, y], axis=0)  -> [3*B, E]
__global__ void pack_seq(const float* __restrict__ state,
                         const float* __restrict__ context,
                         const float* __restrict__ y, float* __restrict__ seq) {
  int i = blockIdx.x * blockDim.x + threadIdx.x;
  if (i >= B_SZ * E_SZ) return;
  seq[i] = state[i];
  seq[B_SZ * E_SZ + i] = context[i];
  seq[2 * B_SZ * E_SZ + i] = y[i];
}

// One GRU step (Keras reset_after=True). Xt = x@gk + bi for this step.
// R = h_prev@grk + br (nullptr on step 0 -> use br broadcast, h_prev = 0).
__global__ void gru_step(const float* __restrict__ Xt, const float* __restrict__ R,
                         const float* __restrict__ br,
                         const float* __restrict__ hprev, float* __restrict__ hout) {
  int i = blockIdx.x * blockDim.x + threadIdx.x;
  if (i >= B_SZ * E_SZ) return;
  const int b = i / E_SZ, e = i % E_SZ;
  const size_t base = (size_t)b * 3 * E_SZ;
  float xz = Xt[base + e], xr = Xt[base + E_SZ + e], xh = Xt[base + 2 * E_SZ + e];
  float rz, rr, rh;
  if (R) {
    rz = R[base + e]; rr = R[base + E_SZ + e]; rh = R[base + 2 * E_SZ + e];
  } else {
    rz = br[e]; rr = br[E_SZ + e]; rh = br[2 * E_SZ + e];
  }
  float h = hprev ? hprev[i] : 0.0f;
  float z = 1.0f / (1.0f + __expf(-(xz + rz)));
  float r = 1.0f / (1.0f + __expf(-(xr + rr)));
  float hh = tanhf(xh + r * rh);
  hout[i] = z * h + (1.0f - z) * hh;
}

// g = concat([y, context, new_state], axis=-1)  -> [B, 3E]
__global__ void pack_g(const float* __restrict__ y, const float* __restrict__ context,
                       const float* __restrict__ h, float* __restrict__ g) {
  int i = blockIdx.x * blockDim.x + threadIdx.x;
  if (i >= B_SZ * E_SZ) return;
  const int b = i / E_SZ, e = i % E_SZ;
  const size_t base = (size_t)b * 3 * E_SZ;
  g[base + e] = y[i];
  g[base + E_SZ + e] = context[i];
  g[base + 2 * E_SZ + e] = h[i];
}

// In-place row softmax over N columns, one block per row.
__global__ __launch_bounds__(256) void softmax_rows(float* __restrict__ X, int N) {
  float* row = X + (size_t)blockIdx.x * N;
  const int lane = threadIdx.x & 31;
  const int wave = threadIdx.x >> 5;
  __shared__ float sm[8];

  float mx = -INFINITY;
  for (int i = threadIdx.x; i < N; i += 256) mx = fmaxf(mx, row[i]);
#pragma unroll
  for (int o = 16; o; o >>= 1) mx = fmaxf(mx, __shfl_xor(mx, o, 32));
  if (lane == 0) sm[wave] = mx;
  __syncthreads();
  float m = sm[0];
#pragma unroll
  for (int i = 1; i < 8; ++i) m = fmaxf(m, sm[i]);
  __syncthreads();

  float s = 0.0f;
  for (int i = threadIdx.x; i < N; i += 256) {
    float e = __expf(row[i] - m);
    row[i] = e;
    s += e;
  }
#pragma unroll
  for (int o = 16; o; o >>= 1) s += __shfl_xor(s, o, 32);
  if (lane == 0) sm[wave] = s;
  __syncthreads();
  float tot = 0.0f;
#pragma unroll
  for (int i = 0; i < 8; ++i) tot += sm[i];
  float inv = 1.0f / tot;
  for (int i = threadIdx.x; i < N; i += 256) row[i] *= inv;
}

// ---------------------------------------------------------------------------
extern "C" void kernel_launch(void* const* d_in, const int* in_sizes, int n_in,
                              void* d_out, int out_size, void* d_ws, size_t ws_size,
                              hipStream_t stream) {
  const float* y      = (const float*)d_in[0];
  const float* state  = (const float*)d_in[1];
  const float* encode = (const float*)d_in[2];
  const float* aW1    = (const float*)d_in[3];
  const float* aW2    = (const float*)d_in[4];
  const float* aW3    = (const float*)d_in[5];
  const float* gk     = (const float*)d_in[6];
  const float* grk    = (const float*)d_in[7];
  const float* gb     = (const float*)d_in[8];
  const float* w1     = (const float*)d_in[9];
  const float* b1     = (const float*)d_in[10];
  const float* w2     = (const float*)d_in[11];
  const float* b2     = (const float*)d_in[12];
  const float* w3     = (const float*)d_in[13];
  const float* b3     = (const float*)d_in[14];
  (void)in_sizes; (void)n_in; (void)out_size; (void)ws_size;

  const float* bi = gb;                // gb[0]
  const float* br = gb + 3 * E_SZ;     // gb[1]

  float* out    = (float*)d_out;
  float* gout   = out;                                   // [B, V]
  float* nstate = out + (size_t)B_SZ * V_SZ;             // [B, E]

  // workspace carve-up (floats)
  float* ws = (float*)d_ws;
  size_t off = 0;
  float* context = ws + off; off += (size_t)B_SZ * E_SZ;
  float* seq     = ws + off; off += (size_t)3 * B_SZ * E_SZ;
  float* X       = ws + off; off += (size_t)3 * B_SZ * 3 * E_SZ;
  float* R       = ws + off; off += (size_t)B_SZ * 3 * E_SZ;
  float* h1      = ws + off; off += (size_t)B_SZ * E_SZ;
  float* h2      = ws + off; off += (size_t)B_SZ * E_SZ;
  float* g       = ws + off; off += (size_t)B_SZ * 3 * E_SZ;
  float* hd1     = ws + off; off += (size_t)B_SZ * H_SZ;
  float* hd2     = ws + off; off += (size_t)B_SZ * H_SZ;

  const int elems = B_SZ * E_SZ;
  const int eb = (elems + 255) / 256;

  // 1) attention -> context
  attention_kernel<<<B_SZ, 256, 0, stream>>>(state, encode, aW1, aW2, aW3, context);

  // 2) GRU: batched input projection for all 3 timesteps, then 3 steps
  pack_seq<<<eb, 256, 0, stream>>>(state, context, y, seq);
  launch_gemm(seq, gk, bi, X, 3 * B_SZ, 3 * E_SZ, E_SZ, 0, stream);     // X = seq@gk + bi
  gru_step<<<eb, 256, 0, stream>>>(X, nullptr, br, nullptr, h1);         // t=0 (h0=0)
  launch_gemm(h1, grk, br, R, B_SZ, 3 * E_SZ, E_SZ, 0, stream);          // R = h1@grk + br
  gru_step<<<eb, 256, 0, stream>>>(X + (size_t)B_SZ * 3 * E_SZ, R, br, h1, h2);
  launch_gemm(h2, grk, br, R, B_SZ, 3 * E_SZ, E_SZ, 0, stream);          // R = h2@grk + br
  gru_step<<<eb, 256, 0, stream>>>(X + (size_t)2 * B_SZ * 3 * E_SZ, R, br, h2, nstate);

  // 3) dense stack
  pack_g<<<eb, 256, 0, stream>>>(y, context, nstate, g);
  launch_gemm(g,   w1, b1, hd1,  B_SZ, H_SZ, 3 * E_SZ, 1, stream);       // tanh
  launch_gemm(hd1, w2, b2, hd2,  B_SZ, H_SZ, H_SZ,     1, stream);       // tanh
  launch_gemm(hd2, w3, b3, gout, B_SZ, V_SZ, H_SZ,     0, stream);       // logits
  softmax_rows<<<B_SZ, 256, 0, stream>>>(gout, V_SZ);                    // in place
}